// ROI_Pool_86260123174470
// MI455X (gfx1250) — compile-verified
//
#include <hip/hip_runtime.h>
#include <stdint.h>

// ROI max-pool for MI455X (gfx1250).
// One workgroup per (roi, pair of 16-channel blocks). Each (rW x rH x 16ch)
// strided tile is fetched global->LDS with a Tensor Data Mover descriptor
// (tensor_load_to_lds, TENSORcnt-tracked). Two tiles are double-buffered so
// the second DMA overlaps compute on the first. 256 threads then compute the
// 16x7x7 max-pooled outputs per tile from LDS.

namespace {
constexpr int C_    = 256;
constexpr int H_    = 224;
constexpr int W_    = 224;
constexpr int PH_   = 7;
constexpr int PW_   = 7;
constexpr int MAXR_ = 25;
constexpr int CB_   = 16;    // channels per TDM tile
constexpr int NBUF_ = 2;     // double buffer
constexpr int THREADS = 256; // 8 wave32

typedef uint32_t v4u __attribute__((ext_vector_type(4)));
typedef uint32_t v8u __attribute__((ext_vector_type(8)));
} // namespace

// Issue one TDM descriptor: load [CB_][rH][rW] f32 tile (strides 1, W, H*W
// elements) from global gaddr into contiguous LDS at lds_addr.
// All arguments are wave-uniform. Tracked by TENSORcnt.
__device__ __forceinline__ void tdm_load_tile(uint32_t lds_addr, uint64_t gaddr,
                                              int rH, int rW) {
    const uint32_t hw = (uint32_t)(H_ * W_);       // 50176

    // D# group 0 (128b): count=1 | lds_addr | global_addr[56:0] | type=2
    v4u g0;
    g0[0] = 1u;                                     // count=1, user mode
    g0[1] = lds_addr;                               // LDS dest (bytes)
    g0[2] = (uint32_t)gaddr;                        // global_addr[31:0]
    g0[3] = ((uint32_t)(gaddr >> 32) & 0x01FFFFFFu) // global_addr[56:32]
          | 0x80000000u;                            // type=2 ("image")

    // D# group 1 (256b)
    v8u g1;
    g1[0] = 2u << 16;                               // data_size=4B, no mask/flags
    g1[1] = ((uint32_t)rW) << 16;                   // tensor_dim0[15:0] (abar=0)
    g1[2] = ((uint32_t)rH) << 16;                   // dim0 hi=0 | tensor_dim1[15:0]
    g1[3] = ((uint32_t)rW) << 16;                   // dim1 hi=0 | tile_dim0=rW
    g1[4] = ((uint32_t)CB_ << 16) | (uint32_t)rH;   // tile_dim1=rH | tile_dim2=CB
    g1[5] = (uint32_t)W_;                           // tensor_dim0_stride[31:0]=W
    g1[6] = (hw & 0xFFFFu) << 16;                   // d0str hi=0 | d1str[15:0]=H*W
    g1[7] = hw >> 16;                               // d1str[47:16]

    // D# group 2 (128b)
    v4u g2;
    g2[0] = (uint32_t)CB_;                          // tensor_dim2 = CB
    g2[1] = 0u;                                     // tensor_dim3 (unused)
    g2[2] = hw;                                     // tensor_dim2_stride = H*W
    g2[3] = 0u;                                     // d2str hi=0 | tile_dim3=0

    // D# group 3 (128b): no dim3/dim4
    v4u g3 = {0u, 0u, 0u, 0u};

    asm volatile("tensor_load_to_lds %0, %1, %2, %3"
                 :: "s"(g0), "s"(g1), "s"(g2), "s"(g3)
                 : "memory");
}

// Max-pool one CB_-channel tile held contiguously in LDS.
__device__ __forceinline__ void pool_tile(const float* __restrict__ tile,
                                          float* __restrict__ out,
                                          int n, int ch0, int rH, int rW) {
    const int area = rH * rW;
    for (int t = threadIdx.x; t < CB_ * PH_ * PW_; t += THREADS) {
        const int c  = t / (PH_ * PW_);
        const int ij = t - c * (PH_ * PW_);
        const int i  = ij / PW_;
        const int j  = ij - i * PW_;
        // reference bins: [i*rH//7, ceil((i+1)*rH/7)) — never empty
        const int rs = (i * rH) / PH_;
        const int re = ((i + 1) * rH + PH_ - 1) / PH_;
        const int cs = (j * rW) / PW_;
        const int ce = ((j + 1) * rW + PW_ - 1) / PW_;

        const float* b = tile + c * area;
        float m = -__builtin_inff();
        for (int r = rs; r < re; ++r) {
            const float* row = b + r * rW;
            for (int q = cs; q < ce; ++q)
                m = fmaxf(m, row[q]);
        }
        out[(((size_t)n * C_ + (ch0 + c)) * PH_ + i) * PW_ + j] = m;
    }
}

__global__ __launch_bounds__(THREADS)
void roi_pool_tdm_kernel(const float* __restrict__ fmap,
                         const int*   __restrict__ rois,
                         float*       __restrict__ out,
                         int nrois) {
    // Two tiles, each [CB_][rH*rW] packed contiguously by the TDM. 80 KB.
    __shared__ __align__(16) float region[NBUF_][CB_ * MAXR_ * MAXR_];

    const int n   = blockIdx.x;                 // roi index
    const int ch0 = blockIdx.y * (CB_ * NBUF_); // first channel of this pair
    if (n >= nrois) return;                     // uniform per block

    const int y  = rois[n * 4 + 0];
    const int x  = rois[n * 4 + 1];
    const int rH = rois[n * 4 + 2];             // odd, 7..25
    const int rW = rois[n * 4 + 3];
    const int y0 = y - (rH - 1) / 2;            // in-bounds by setup clip
    const int x0 = x - (rW - 1) / 2;

    const size_t tile_off = ((size_t)ch0 * H_ + (size_t)y0) * W_ + x0;

    // ---- Phase 1: wave 0 issues both DMAs, waits for the first only ----
    if ((threadIdx.x >> 5) == 0) {
        const uint64_t ga0 = (uint64_t)(uintptr_t)(fmap + tile_off);
        const uint64_t ga1 = (uint64_t)(uintptr_t)(fmap + tile_off
                                                   + (size_t)CB_ * H_ * W_);
        tdm_load_tile((uint32_t)(uintptr_t)(&region[0][0]), ga0, rH, rW);
        tdm_load_tile((uint32_t)(uintptr_t)(&region[1][0]), ga1, rH, rW);
        // TDM ops from one wave complete in order: cnt<=1 => first tile done.
        __builtin_amdgcn_s_wait_tensorcnt(1);
    }
    __syncthreads();

    // ---- Phase 2a: pool tile 0 while tile 1's DMA is still in flight ----
    pool_tile(&region[0][0], out, n, ch0, rH, rW);

    // ---- Phase 2b: wait for tile 1, then pool it ----
    if ((threadIdx.x >> 5) == 0) {
        __builtin_amdgcn_s_wait_tensorcnt(0);
    }
    __syncthreads();
    pool_tile(&region[1][0], out, n, ch0 + CB_, rH, rW);
}

extern "C" void kernel_launch(void* const* d_in, const int* in_sizes, int n_in,
                              void* d_out, int out_size, void* d_ws, size_t ws_size,
                              hipStream_t stream) {
    const float* fmap = (const float*)d_in[0];   // (256, 224, 224) f32
    const int*   rois = (const int*)d_in[1];     // (N, 4) i32
    float*       out  = (float*)d_out;           // (N, 256, 7, 7) f32

    const int nrois = in_sizes[1] / 4;
    dim3 grid(nrois, C_ / (CB_ * NBUF_), 1);     // (512, 8)
    roi_pool_tdm_kernel<<<grid, THREADS, 0, stream>>>(fmap, rois, out, nrois);
}